// GAT_38474317038201
// MI455X (gfx1250) — compile-verified
//
#include <hip/hip_runtime.h>
#include <math.h>

// ---------------- problem constants ----------------
#define HEADS      4
#define HCF        128     // hidden = HEADS*32
#define C2         47      // output channels per head, layer 2
#define HC2        188     // HEADS*47
#define HC2P       192     // padded
#define SK2P       64      // padded skip width for layer2
#define NEG_SLOPE  0.2f
#define EPS_DEN    1e-16f

typedef float v2f __attribute__((ext_vector_type(2)));
typedef float v8f __attribute__((ext_vector_type(8)));

// monotone uint key for float atomic max
__device__ __forceinline__ unsigned fkey(float f) {
  unsigned u = __float_as_uint(f);
  return (u & 0x80000000u) ? ~u : (u | 0x80000000u);
}
__device__ __forceinline__ float fdecode(unsigned k) {
  unsigned u = (k & 0x80000000u) ? (k ^ 0x80000000u) : ~k;
  return __uint_as_float(u);
}

// ---------------- fp32 WMMA GEMM, LDS-staged B -------------------
// C[M x LDC] = A[M x K] * B[K x LDC], all leading dims compile-time so
// every strided access folds into instruction-immediate offsets.
// Block = 8 waves sharing one 64-col strip of B staged in LDS in
// WMMA-fragment order (each lane's (b.x,b.y) pair = one ds_load_b64).
// Each wave owns one 16-row tile -> 4x V_WMMA_F32_16X16X4_F32 per k-step.
template <int K, int LDA, int LDB, int LDC>
__global__ __launch_bounds__(256)
void gemm_wmma_f32(const float* __restrict__ A, const float* __restrict__ B,
                   float* __restrict__ C, int M) {
  __shared__ float bs[(K / 4) * 128 * 2];     // 32 ks * (4 tiles * 32 lanes) * 2 = 32 KB

  const int tid  = threadIdx.x;
  const int lane = tid & 31;
  const int wave = tid >> 5;                  // 0..7
  const int tn   = blockIdx.y * 64;

  // ---- cooperative stage of the K x 64 strip of B, fragment-ordered ----
  // idx = ks*128 + t*32 + l  ->  pair (B[k][col], B[k+1][col])
  for (int idx = tid; idx < (K / 4) * 128; idx += 256) {
    const int l   = idx & 31;
    const int t   = (idx >> 5) & 3;
    const int ks  = idx >> 7;
    const int k   = 4 * ks + ((l >> 4) << 1);
    const int col = tn + t * 16 + (l & 15);
    float2 b;
    b.x = B[(size_t)k * LDB + col];
    b.y = B[(size_t)(k + 1) * LDB + col];
    *(float2*)&bs[(size_t)idx * 2] = b;
  }
  __syncthreads();

  const int rlo   = lane & 15;
  const int khalf = (lane >> 4) << 1;         // 0 or 2
  const int tm    = blockIdx.x * 8 + wave;

  int arow = tm * 16 + rlo;
  if (arow >= M) arow = M - 1;                // clamp (stores guarded)
  const float* ap = A + (size_t)arow * LDA;

  v8f acc[4] = {};
#pragma unroll 4
  for (int ks = 0; ks < K / 4; ++ks) {
    v2f a;
    a.x = ap[4 * ks + khalf];
    a.y = ap[4 * ks + khalf + 1];
    const float* bp = &bs[(size_t)(ks * 128 + lane) * 2];
#pragma unroll
    for (int t = 0; t < 4; ++t) {
      v2f b = *(const v2f*)(bp + t * 64);     // ds_load_b64
      acc[t] = __builtin_amdgcn_wmma_f32_16x16x4_f32(
          false, a, false, b, (short)0, acc[t], false, false);
    }
  }

  const int mbase = tm * 16 + ((lane >> 4) << 3);   // M = r (+8 for hi half)
  if ((tm + 1) * 16 <= M) {
    // whole tile in range: flat stores, immediate offsets
    float* cp = C + (size_t)mbase * LDC + tn + rlo;
#pragma unroll
    for (int t = 0; t < 4; ++t)
#pragma unroll
      for (int r = 0; r < 8; ++r)
        cp[r * LDC + t * 16] = acc[t][r];
  } else {
#pragma unroll
    for (int t = 0; t < 4; ++t)
#pragma unroll
      for (int r = 0; r < 8; ++r) {
        const int row = mbase + r;
        if (row < M) C[(size_t)row * LDC + tn + t * 16 + rlo] = acc[t][r];
      }
  }
}

// ---------------- attention coefficients -----------
// asrc[n,h] = dot(xh[n,h,:], att_src[h,:]); likewise adst.
template <int C, int LD>
__global__ void att_kernel(const float* __restrict__ xh,
                           const float* __restrict__ att_src,
                           const float* __restrict__ att_dst,
                           float* __restrict__ asrc, float* __restrict__ adst,
                           int N) {
  int id = blockIdx.x * blockDim.x + threadIdx.x;
  if (id >= N * HEADS) return;
  int n = id >> 2, h = id & 3;
  const float* xp = xh + (size_t)n * LD + h * C;
  const float* ws = att_src + h * C;
  const float* wd = att_dst + h * C;
  float s1 = 0.f, s2 = 0.f;
#pragma unroll 8
  for (int c = 0; c < C; ++c) {
    float v = xp[c];
    s1 += v * ws[c];
    s2 += v * wd[c];
  }
  asrc[id] = s1;
  adst[id] = s2;
}

// ---------------- edge pass 1: e + segment max -----
__global__ void edge_pre(const int* __restrict__ ei, int E0, int EP,
                         const float* __restrict__ asrc,
                         const float* __restrict__ adst,
                         float* __restrict__ eb, unsigned* __restrict__ mkey) {
  int e = blockIdx.x * blockDim.x + threadIdx.x;
  if (e >= EP) return;
  int src = (e < E0) ? ei[e]      : (e - E0);
  int dst = (e < E0) ? ei[E0 + e] : (e - E0);
  const float4 es = *(const float4*)(asrc + (size_t)src * 4);
  const float4 ed = *(const float4*)(adst + (size_t)dst * 4);
  float v[4] = {es.x + ed.x, es.y + ed.y, es.z + ed.z, es.w + ed.w};
#pragma unroll
  for (int h = 0; h < 4; ++h) {
    float x = v[h];
    x = x > 0.f ? x : NEG_SLOPE * x;
    v[h] = x;
    atomicMax(&mkey[(size_t)dst * 4 + h], fkey(x));
  }
  *(float4*)(eb + (size_t)e * 4) = float4{v[0], v[1], v[2], v[3]};
}

// ---------------- edge pass 2: exp + segment sum ---
__global__ void edge_sum(const int* __restrict__ ei, int E0, int EP,
                         const unsigned* __restrict__ mkey,
                         float* __restrict__ eb, float* __restrict__ sden) {
  int e = blockIdx.x * blockDim.x + threadIdx.x;
  if (e >= EP) return;
  int dst = (e < E0) ? ei[E0 + e] : (e - E0);
  float4 ev = *(const float4*)(eb + (size_t)e * 4);
  uint4 mk  = *(const uint4*)(mkey + (size_t)dst * 4);     // global_load_b128
  float v[4] = {ev.x, ev.y, ev.z, ev.w};
  unsigned k[4] = {mk.x, mk.y, mk.z, mk.w};
#pragma unroll
  for (int h = 0; h < 4; ++h) {
    float ex = __expf(v[h] - fdecode(k[h]));
    v[h] = ex;
    atomicAdd(&sden[(size_t)dst * 4 + h], ex);
  }
  *(float4*)(eb + (size_t)e * 4) = float4{v[0], v[1], v[2], v[3]};
}

// ---------------- edge pass 3: weighted aggregate --
// warp per edge: gather xh[src], scale by alpha[h], scatter-add into agg[dst].
template <int C, int LD>
__global__ __launch_bounds__(256)
void edge_agg(const int* __restrict__ ei, int E0, int EP,
              const float* __restrict__ xh, const float* __restrict__ eb,
              const float* __restrict__ sden, float* __restrict__ agg) {
  int e = blockIdx.x * (blockDim.x >> 5) + (threadIdx.x >> 5);
  if (e >= EP) return;
  const int lane = threadIdx.x & 31;
  int src = (e < E0) ? ei[e]      : (e - E0);
  int dst = (e < E0) ? ei[E0 + e] : (e - E0);
  const float* xs = xh + (size_t)src * LD;
  // overlap the gather-row fetch with the alpha loads (global_prefetch_b8)
  __builtin_prefetch(xs + lane * 4, 0, 3);
  const float4 ex4 = *(const float4*)(eb + (size_t)e * 4);     // b128
  const float4 s4  = *(const float4*)(sden + (size_t)dst * 4); // b128
  float alpha[4] = {ex4.x / (s4.x + EPS_DEN), ex4.y / (s4.y + EPS_DEN),
                    ex4.z / (s4.z + EPS_DEN), ex4.w / (s4.w + EPS_DEN)};
  float* ad = agg + (size_t)dst * LD;
#pragma unroll
  for (int base = 0; base < HEADS * C; base += 32) {
    int f = base + lane;
    if (f < HEADS * C) {
      int h = f / C;
      atomicAdd(&ad[f], xs[f] * alpha[h]);
    }
  }
}

// ---------------- epilogues ------------------------
__global__ void combine_elu(const float* __restrict__ agg,
                            const float* __restrict__ skip,
                            const float* __restrict__ bias,
                            const float* __restrict__ skb,
                            float* __restrict__ out, int total) {
  int i = blockIdx.x * blockDim.x + threadIdx.x;
  if (i >= total) return;
  int f = i & 127;
  float v = agg[i] + bias[f] + skip[i] + skb[f];
  out[i] = v > 0.f ? v : expm1f(v);
}

__global__ void combine_out(const float* __restrict__ agg,   // ld HC2P
                            const float* __restrict__ skip,  // ld SK2P
                            const float* __restrict__ bias2,
                            const float* __restrict__ skb2,
                            float* __restrict__ out, int N) {
  int i = blockIdx.x * blockDim.x + threadIdx.x;
  if (i >= N * C2) return;
  int n = i / C2, o = i - n * C2;
  const float* a = agg + (size_t)n * HC2P;
  float m = 0.25f * (a[o] + a[C2 + o] + a[2 * C2 + o] + a[3 * C2 + o]);
  out[i] = m + bias2[o] + skip[(size_t)n * SK2P + o] + skb2[o];
}

__global__ void pad_w(const float* __restrict__ w, float* __restrict__ wp,
                      int rows, int cols, int ldp) {
  int i = blockIdx.x * blockDim.x + threadIdx.x;
  if (i >= rows * ldp) return;
  int r = i / ldp, c = i - r * ldp;
  wp[i] = (c < cols) ? w[(size_t)r * cols + c] : 0.f;
}

// ---------------- driver ---------------------------
extern "C" void kernel_launch(void* const* d_in, const int* in_sizes, int n_in,
                              void* d_out, int out_size, void* d_ws, size_t ws_size,
                              hipStream_t stream) {
  const float* x   = (const float*)d_in[0];
  const int*   ei  = (const int*)d_in[1];
  const float* lw[3] = {(const float*)d_in[2], (const float*)d_in[8],  (const float*)d_in[14]};
  const float* as[3] = {(const float*)d_in[3], (const float*)d_in[9],  (const float*)d_in[15]};
  const float* ad[3] = {(const float*)d_in[4], (const float*)d_in[10], (const float*)d_in[16]};
  const float* bi[3] = {(const float*)d_in[5], (const float*)d_in[11], (const float*)d_in[17]};
  const float* sw[3] = {(const float*)d_in[6], (const float*)d_in[12], (const float*)d_in[18]};
  const float* sb[3] = {(const float*)d_in[7], (const float*)d_in[13], (const float*)d_in[19]};

  const int N  = in_sizes[0] / 128;
  const int E0 = in_sizes[1] / 2;
  const int EP = E0 + N;

  // workspace carve-up (floats)
  float* w = (float*)d_ws;
  float*    xcur = w;                      w += (size_t)N * HCF;
  float*    xh   = w;                      w += (size_t)N * HC2P;
  float*    agg  = w;                      w += (size_t)N * HC2P;
  float*    skip = w;                      w += (size_t)N * HCF;
  float*    asrc = w;                      w += (size_t)N * 4;
  float*    adst = w;                      w += (size_t)N * 4;
  unsigned* mkey = (unsigned*)w;           w += (size_t)N * 4;
  float*    sden = w;                      w += (size_t)N * 4;
  float*    eb   = w;                      w += (size_t)EP * 4;
  float*    w2p  = w;                      w += (size_t)128 * HC2P;
  float*    sw2p = w;                      w += (size_t)128 * SK2P;

  const int mt8  = (N + 127) / 128;        // blocks of 8 16-row tiles
  const int eblk = (EP + 255) / 256;
  const int ewarp = (EP + 7) / 8;          // 8 edges per 256-thread block

  // ================= layer 0 (128 -> 128) =================
  gemm_wmma_f32<128, 128, 128, 128><<<dim3(mt8, 2), 256, 0, stream>>>(x, lw[0], xh,   N);
  gemm_wmma_f32<128, 128, 128, 128><<<dim3(mt8, 2), 256, 0, stream>>>(x, sw[0], skip, N);
  att_kernel<32, 128><<<(N * 4 + 255) / 256, 256, 0, stream>>>(xh, as[0], ad[0], asrc, adst, N);
  hipMemsetAsync(mkey, 0, (size_t)N * 4 * sizeof(unsigned), stream);
  hipMemsetAsync(sden, 0, (size_t)N * 4 * sizeof(float), stream);
  hipMemsetAsync(agg,  0, (size_t)N * HCF * sizeof(float), stream);
  edge_pre<<<eblk, 256, 0, stream>>>(ei, E0, EP, asrc, adst, eb, mkey);
  edge_sum<<<eblk, 256, 0, stream>>>(ei, E0, EP, mkey, eb, sden);
  edge_agg<32, 128><<<ewarp, 256, 0, stream>>>(ei, E0, EP, xh, eb, sden, agg);
  combine_elu<<<((size_t)N * HCF + 255) / 256, 256, 0, stream>>>(agg, skip, bi[0], sb[0], xcur, N * HCF);

  // ================= layer 1 (128 -> 128) =================
  gemm_wmma_f32<128, 128, 128, 128><<<dim3(mt8, 2), 256, 0, stream>>>(xcur, lw[1], xh,   N);
  gemm_wmma_f32<128, 128, 128, 128><<<dim3(mt8, 2), 256, 0, stream>>>(xcur, sw[1], skip, N);
  att_kernel<32, 128><<<(N * 4 + 255) / 256, 256, 0, stream>>>(xh, as[1], ad[1], asrc, adst, N);
  hipMemsetAsync(mkey, 0, (size_t)N * 4 * sizeof(unsigned), stream);
  hipMemsetAsync(sden, 0, (size_t)N * 4 * sizeof(float), stream);
  hipMemsetAsync(agg,  0, (size_t)N * HCF * sizeof(float), stream);
  edge_pre<<<eblk, 256, 0, stream>>>(ei, E0, EP, asrc, adst, eb, mkey);
  edge_sum<<<eblk, 256, 0, stream>>>(ei, E0, EP, mkey, eb, sden);
  edge_agg<32, 128><<<ewarp, 256, 0, stream>>>(ei, E0, EP, xh, eb, sden, agg);
  combine_elu<<<((size_t)N * HCF + 255) / 256, 256, 0, stream>>>(agg, skip, bi[1], sb[1], xcur, N * HCF);

  // ================= layer 2 (128 -> 47, head mean) =======
  pad_w<<<(128 * HC2P + 255) / 256, 256, 0, stream>>>(lw[2], w2p,  128, HC2, HC2P);
  pad_w<<<(128 * SK2P + 255) / 256, 256, 0, stream>>>(sw[2], sw2p, 128, C2,  SK2P);
  gemm_wmma_f32<128, 128, HC2P, HC2P><<<dim3(mt8, 3), 256, 0, stream>>>(xcur, w2p,  xh,   N);
  gemm_wmma_f32<128, 128, SK2P, SK2P><<<dim3(mt8, 1), 256, 0, stream>>>(xcur, sw2p, skip, N);
  att_kernel<47, HC2P><<<(N * 4 + 255) / 256, 256, 0, stream>>>(xh, as[2], ad[2], asrc, adst, N);
  hipMemsetAsync(mkey, 0, (size_t)N * 4 * sizeof(unsigned), stream);
  hipMemsetAsync(sden, 0, (size_t)N * 4 * sizeof(float), stream);
  hipMemsetAsync(agg,  0, (size_t)N * HC2P * sizeof(float), stream);
  edge_pre<<<eblk, 256, 0, stream>>>(ei, E0, EP, asrc, adst, eb, mkey);
  edge_sum<<<eblk, 256, 0, stream>>>(ei, E0, EP, mkey, eb, sden);
  edge_agg<47, HC2P><<<ewarp, 256, 0, stream>>>(ei, E0, EP, xh, eb, sden, agg);
  combine_out<<<(N * C2 + 255) / 256, 256, 0, stream>>>(agg, skip, bi[2], sb[2], (float*)d_out, N);
}